// TempSpatialAttention_1511828488298
// MI455X (gfx1250) — compile-verified
//
#include <hip/hip_runtime.h>

typedef __attribute__((ext_vector_type(2))) float v2f;
typedef __attribute__((ext_vector_type(8))) float v8f;
typedef __attribute__((ext_vector_type(4))) unsigned int v4u;
typedef __attribute__((ext_vector_type(8))) int v8i;
typedef __attribute__((ext_vector_type(4))) int v4i;

#define ATT_SCALE 0.17677669529663687f  // 32^-0.5

// ---- LDS layout (in floats) ----
// k/v planes: [c (32)][x+1 (130)]  (col 0 = x=-1 halo, col 129 = x=128 halo)
#define PLANE_F   (32 * 130)              // 4160
#define K_OFF     0                       // 3 planes (y-1, y, y+1)
#define V_OFF     (3 * PLANE_F)           // 12480
#define Q_OFF     (6 * PLANE_F)           // 24960, layout [c(32)][x(128)]
#define SM_OFF    (Q_OFF + 32 * 128)      // 29056, per-wave 16x17 score scratch
#define FX_OFF    (SM_OFF + 8 * 16 * 17)  // 31232, per-wave 12-float fixups
#define SMEM_F    (FX_OFF + 8 * 12)       // 31328 floats = 125312 B

__launch_bounds__(256)
__global__ void na3x3_kernel(const float* __restrict__ qg,
                             const float* __restrict__ kg,
                             const float* __restrict__ vg,
                             float* __restrict__ outg)
{
    __shared__ float smem[SMEM_F];

    const int tid  = threadIdx.x;
    const int lane = tid & 31;
    const int wave = tid >> 5;
    const int bid  = blockIdx.x;
    const int y = bid & 127;
    const int h = (bid >> 7) & 3;
    const int b = bid >> 9;
    const int x0 = wave * 16;     // 16-pixel segment per wave
    const int lh = lane >> 4;     // lane half (selects K pair / channel group)
    const int p  = lane & 15;     // pixel within segment (M / N index)

    // ---- zero whole planes for out-of-range y rows (y=0 / y=127 borders) ----
    #pragma unroll
    for (int pl = 0; pl < 3; ++pl) {
        int yy = y - 1 + pl;
        if (yy < 0 || yy > 127) {
            for (int i = tid; i < PLANE_F; i += 256) {
                smem[K_OFF + pl * PLANE_F + i] = 0.f;
                smem[V_OFF + pl * PLANE_F + i] = 0.f;
            }
        }
    }
    // ---- zero x-halo cells (cols 0 and 129 of every c-row, all 6 planes) ----
    for (int i = tid; i < 6 * 64; i += 256) {
        int pl = i >> 6, cell = i & 63;
        int c = cell >> 1, side = cell & 1;
        int base = (pl < 3) ? (K_OFF + pl * PLANE_F) : (V_OFF + (pl - 3) * PLANE_F);
        smem[base + c * 130 + side * 129] = 0.f;
    }

    // ---- TDM async tile loads: waves 0-2 -> k rows, 3-5 -> v rows, 6 -> q row ----
    // Each plane: dim0 = 128 floats along x (contiguous), 32 c-rows (stride 16384),
    // LDS pad of 2 dwords after every 128 dwords => 130-float pitch with halo slots.
    if (wave < 7) {
        bool isq = (wave == 6);
        int tdy  = isq ? 0 : (wave % 3);
        bool isv = (!isq) && (wave >= 3);
        int yy   = isq ? y : (y - 1 + tdy);
        if (yy >= 0 && yy <= 127) {
            const float* gp = isq ? qg : (isv ? vg : kg);
            unsigned long long ga = (unsigned long long)(const void*)(gp
                + (unsigned long long)(b * 128 + h * 32) * 16384ull
                + (unsigned long long)yy * 128ull);
            unsigned ldsb = isq ? (unsigned)(Q_OFF * 4)
                                : (unsigned)(((isv ? V_OFF : K_OFF) + tdy * PLANE_F) * 4 + 4);
            v4u g0;
            g0[0] = 1u;                                   // count = 1 (valid D#)
            g0[1] = ldsb;                                 // lds_addr
            g0[2] = (unsigned)(ga & 0xffffffffull);       // global_addr lo
            g0[3] = (unsigned)(ga >> 32) | (2u << 30);    // global_addr hi | type=2
            v8i g1;
            g1[0] = isq ? (2 << 16)                                          // data_size=4B
                        : ((2 << 16) | (1 << 20) | (6 << 22) | (1 << 25));   // + pad 2dw/128dw
            g1[1] = (128 << 16);   // tensor_dim0[15:0] << 16
            g1[2] = (32 << 16);    // tensor_dim0 hi=0 | tensor_dim1[15:0]<<16
            g1[3] = (128 << 16);   // tensor_dim1 hi=0 | tile_dim0<<16
            g1[4] = 32;            // tile_dim1 | tile_dim2=0
            g1[5] = 16384;         // tensor_dim0_stride lo32
            g1[6] = 0;             // stride0 hi | stride1 lo
            g1[7] = 0;
            v4i g2 = {0, 0, 0, 0};
            v4i g3 = {0, 0, 0, 0};
            v8i g4 = {0, 0, 0, 0, 0, 0, 0, 0};
            __builtin_amdgcn_tensor_load_to_lds(g0, g1, g2, g3, g4, 0);
            __builtin_amdgcn_s_wait_tensorcnt(0);
        }
    }
    __syncthreads();

    // ---- QK^T via chained V_WMMA_F32_16X16X4_F32 (exact fp32) ----
    // A (16x4): lane<16: M=lane, K pair (0,1); lane>=16: M=lane-16, K pair (2,3)
    const float* qpl = &smem[Q_OFF];
    v2f a[8];
    #pragma unroll
    for (int kk = 0; kk < 8; ++kk) {
        int c = 4 * kk + 2 * lh;
        a[kk][0] = qpl[c * 128 + x0 + p];
        a[kk][1] = qpl[(c + 1) * 128 + x0 + p];
    }

    float s[9];
    float* smw = &smem[SM_OFF + wave * (16 * 17)];
    #pragma unroll
    for (int dy = 0; dy < 3; ++dy) {
        const float* kpl = &smem[K_OFF + dy * PLANE_F];
        v8f S = {0.f, 0.f, 0.f, 0.f, 0.f, 0.f, 0.f, 0.f};
        #pragma unroll
        for (int kk = 0; kk < 8; ++kk) {
            int c = 4 * kk + 2 * lh;
            // B (4x16): lane = window column n; window col n -> pixel x0-1+n -> LDS col x0+n
            v2f bb;
            bb[0] = kpl[c * 130 + x0 + p];
            bb[1] = kpl[(c + 1) * 130 + x0 + p];
            S = __builtin_amdgcn_wmma_f32_16x16x4_f32(
                    false, a[kk], false, bb, (short)0, S, false, false);
        }
        // scatter C (lane<16: N=lane, M=r; lane>=16: N=lane-16, M=r+8) to LDS
        int mb = lh * 8;
        #pragma unroll
        for (int r = 0; r < 8; ++r) smw[(mb + r) * 17 + p] = S[r];
        // band gather: score(pixel p, dx=dxi-1) lives at S[p][p+dxi]
        #pragma unroll
        for (int dxi = 0; dxi < 3; ++dxi) {
            int n = p + dxi;
            s[3 * dy + dxi] = (n < 16) ? smw[p * 17 + n] : 0.f;
        }
    }

    // ---- fixups: the 3 band entries per dy outside the 16-col window ----
    float* fxw = &smem[FX_OFF + wave * 12];
    if (lane < 9) {
        int fdy = lane / 3, t = lane % 3;
        int m  = (t == 0) ? 14 : 15;
        int dx = (t == 1) ? 0 : 1;
        const float* kpl = &smem[K_OFF + fdy * PLANE_F];
        int col = x0 + m + dx + 1;     // neighbor pixel + halo shift
        float acc = 0.f;
        #pragma unroll
        for (int c = 0; c < 32; ++c)
            acc = __fmaf_rn(qpl[c * 128 + x0 + m], kpl[c * 130 + col], acc);
        fxw[lane] = acc;
    }
    if (p == 14) {
        #pragma unroll
        for (int d2 = 0; d2 < 3; ++d2) s[3 * d2 + 2] = fxw[3 * d2 + 0];
    }
    if (p == 15) {
        #pragma unroll
        for (int d2 = 0; d2 < 3; ++d2) {
            s[3 * d2 + 1] = fxw[3 * d2 + 1];
            s[3 * d2 + 2] = fxw[3 * d2 + 2];
        }
    }

    // ---- softmax over 9 (scale then softmax; zero-padded neighbors included) ----
    float mx = s[0];
    #pragma unroll
    for (int j = 1; j < 9; ++j) mx = fmaxf(mx, s[j]);
    float e[9], sum = 0.f;
    #pragma unroll
    for (int j = 0; j < 9; ++j) { e[j] = __expf((s[j] - mx) * ATT_SCALE); sum += e[j]; }
    float inv = 1.f / sum;

    // ---- AV: lane handles pixel p, channels [lh*16, lh*16+16) ----
    float acc[16];
    #pragma unroll
    for (int i = 0; i < 16; ++i) acc[i] = 0.f;
    int coff = lh * 16;
    #pragma unroll
    for (int j = 0; j < 9; ++j) {
        int d2 = j / 3, dxi = j % 3;
        const float* vpl = &smem[V_OFF + d2 * PLANE_F];
        int col = x0 + p + dxi;        // pixel p+dx-1 -> LDS col p+dx (halo shift)
        float wj = e[j] * inv;
        #pragma unroll
        for (int i = 0; i < 16; ++i)
            acc[i] = __fmaf_rn(wj, vpl[(coff + i) * 130 + col], acc[i]);
    }

    // ---- ReLU + store: out[b][y][x][h*32+c], 4x float4 per lane ----
    size_t obase = (((size_t)b * 128 + y) * 128 + (x0 + p)) * 128 + h * 32 + coff;
    float4* ov = reinterpret_cast<float4*>(outg + obase);
    #pragma unroll
    for (int i4 = 0; i4 < 4; ++i4) {
        float4 t4;
        t4.x = fmaxf(acc[4 * i4 + 0], 0.f);
        t4.y = fmaxf(acc[4 * i4 + 1], 0.f);
        t4.z = fmaxf(acc[4 * i4 + 2], 0.f);
        t4.w = fmaxf(acc[4 * i4 + 3], 0.f);
        ov[i4] = t4;
    }
}

extern "C" void kernel_launch(void* const* d_in, const int* in_sizes, int n_in,
                              void* d_out, int out_size, void* d_ws, size_t ws_size,
                              hipStream_t stream) {
    const float* q = (const float*)d_in[0];
    const float* k = (const float*)d_in[1];
    const float* v = (const float*)d_in[2];
    float* out = (float*)d_out;
    // one block per (b, h, y): 4*4*128 = 2048 blocks, 256 threads = 8 waves x 16 px
    na3x3_kernel<<<dim3(2048), dim3(256), 0, stream>>>(q, k, v, out);
}